// Attn_51616916963552
// MI455X (gfx1250) — compile-verified
//
#include <hip/hip_runtime.h>
#include <hip/hip_bf16.h>
#include <math.h>

// ---------------------------------------------------------------------------
// attn[s] = softmax_s( enc[s,:] . v ),  v[h] = sum_o hidden[o] * W[o,h]
// (b contributes only a uniform shift -> dropped; softmax invariant)
// ---------------------------------------------------------------------------

#define HDIM 1024
#define SDIM 32768

typedef __attribute__((ext_vector_type(2))) float v2f;
typedef __attribute__((ext_vector_type(8))) float v8f;

// Workspace layout (bytes):
//   [0 .. 4096)            : v  (1024 f32)
//   [4096 .. 135168)       : energies (32768 f32)
//   [135168 .. 137216)     : per-block max (512 f32)
//   [137216 .. 137220)     : global max (1 f32)
//   [137728 .. 138240)     : per-block sums (128 f32)
#define WS_V_OFF     0
#define WS_E_OFF     4096
#define WS_BMAX_OFF  135168
#define WS_GMAX_OFF  137216
#define WS_BSUM_OFF  137728

#define NB_ENERGY 512   // blocks for the big GEMV (8 waves * 8 rows each)
#define NB_SOFT   128   // blocks for exp/scale (256 threads -> 32768 lanes)

// ---------------------------------------------------------------------------
// Kernel 1: v = hidden @ W  via V_WMMA_F32_16X16X4_F32.
// One wave (32 threads) per block; block n computes v[16n .. 16n+15].
// A (16x4, f32): every row m = hidden[k0..k0+3] (broadcast over M).
//   VGPR0: lanes0-15 K=k0,   lanes16-31 K=k0+2
//   VGPR1: lanes0-15 K=k0+1, lanes16-31 K=k0+3
// B (4x16, f32): B[k, n] = W[(k0+k)*H + n0+n]
//   VGPR0: lanes0-15 row K=k0,   lanes16-31 row K=k0+2  (N = lane%16)
//   VGPR1: lanes0-15 row K=k0+1, lanes16-31 row K=k0+3
// D: VGPR0, lanes0-15 = (M=0, N=lane) -> v[n0+lane].
// ---------------------------------------------------------------------------
__global__ __launch_bounds__(32)
void k_v_wmma(const float* __restrict__ hidden,
              const float* __restrict__ W,
              float* __restrict__ v_out) {
    const int lane = threadIdx.x;           // 0..31, full wave => EXEC all 1s
    const int half = lane >> 4;             // 0 for lanes 0-15, 1 for 16-31
    const int nn   = lane & 15;
    const int n0   = blockIdx.x * 16;

    v8f acc = {};
    #pragma unroll 4
    for (int k0 = 0; k0 < HDIM; k0 += 4) {
        const int ka = k0 + 2 * half;
        v2f a, b;
        a.x = hidden[ka];
        a.y = hidden[ka + 1];
        b.x = W[(size_t)ka * HDIM + n0 + nn];
        b.y = W[(size_t)(ka + 1) * HDIM + n0 + nn];
        acc = __builtin_amdgcn_wmma_f32_16x16x4_f32(
            /*neg_a=*/false, a, /*neg_b=*/false, b,
            /*c_mod=*/(short)0, acc, /*reuse_a=*/false, /*reuse_b=*/false);
    }
    // All M rows identical; take M=0 from VGPR0 on lanes 0-15.
    if (lane < 16) v_out[n0 + lane] = acc[0];
}

// ---------------------------------------------------------------------------
// Kernel 2: e[s] = enc[s,:] . v   (bandwidth-bound GEMV, 128 MB stream)
// 512 blocks * 256 threads (8 waves). Each wave: 8 rows (2 batches of 4).
// v staged in LDS (ds_load_b128). Coalesced float4 loads: lane l covers
// h = it*128 + 4l -> 512B contiguous per load instruction per row.
// ---------------------------------------------------------------------------
__global__ __launch_bounds__(256)
void k_energy(const float* __restrict__ enc,
              const float* __restrict__ v,
              float* __restrict__ e_out,
              float* __restrict__ bmax) {
    __shared__ float vs[HDIM];
    __shared__ float wred[8];

    const int tid  = threadIdx.x;
    const int lane = tid & 31;
    const int wave = tid >> 5;

    for (int i = tid; i < HDIM; i += 256) vs[i] = v[i];
    __syncthreads();

    const int rowBase = (blockIdx.x * 8 + wave) * 8;   // 8 rows per wave
    float wmax = -INFINITY;

    #pragma unroll
    for (int batch = 0; batch < 2; ++batch) {
        const int r0 = rowBase + batch * 4;
        float acc0 = 0.f, acc1 = 0.f, acc2 = 0.f, acc3 = 0.f;
        #pragma unroll
        for (int it = 0; it < 8; ++it) {
            const int h = it * 128 + lane * 4;
            const float4 vv = *(const float4*)(vs + h);
            const float4 e0 = *(const float4*)(enc + (size_t)(r0 + 0) * HDIM + h);
            const float4 e1 = *(const float4*)(enc + (size_t)(r0 + 1) * HDIM + h);
            const float4 e2 = *(const float4*)(enc + (size_t)(r0 + 2) * HDIM + h);
            const float4 e3 = *(const float4*)(enc + (size_t)(r0 + 3) * HDIM + h);
            acc0 += e0.x*vv.x + e0.y*vv.y + e0.z*vv.z + e0.w*vv.w;
            acc1 += e1.x*vv.x + e1.y*vv.y + e1.z*vv.z + e1.w*vv.w;
            acc2 += e2.x*vv.x + e2.y*vv.y + e2.z*vv.z + e2.w*vv.w;
            acc3 += e3.x*vv.x + e3.y*vv.y + e3.z*vv.z + e3.w*vv.w;
        }
        // wave32 butterfly reduction -> every lane holds the full sums
        #pragma unroll
        for (int off = 16; off > 0; off >>= 1) {
            acc0 += __shfl_xor(acc0, off, 32);
            acc1 += __shfl_xor(acc1, off, 32);
            acc2 += __shfl_xor(acc2, off, 32);
            acc3 += __shfl_xor(acc3, off, 32);
        }
        if (lane == 0) {
            e_out[r0 + 0] = acc0;
            e_out[r0 + 1] = acc1;
            e_out[r0 + 2] = acc2;
            e_out[r0 + 3] = acc3;
        }
        wmax = fmaxf(wmax, fmaxf(fmaxf(acc0, acc1), fmaxf(acc2, acc3)));
    }

    if (lane == 0) wred[wave] = wmax;
    __syncthreads();
    if (tid == 0) {
        float m = wred[0];
        #pragma unroll
        for (int i = 1; i < 8; ++i) m = fmaxf(m, wred[i]);
        bmax[blockIdx.x] = m;
    }
}

// ---------------------------------------------------------------------------
// Kernel 3: global max of 512 per-block maxima (single block).
// ---------------------------------------------------------------------------
__global__ __launch_bounds__(256)
void k_gmax(const float* __restrict__ bmax, float* __restrict__ gmax) {
    __shared__ float sd[256];
    const int t = threadIdx.x;
    sd[t] = fmaxf(bmax[t], bmax[t + 256]);
    __syncthreads();
    for (int s = 128; s > 0; s >>= 1) {
        if (t < s) sd[t] = fmaxf(sd[t], sd[t + s]);
        __syncthreads();
    }
    if (t == 0) *gmax = sd[0];
}

// ---------------------------------------------------------------------------
// Kernel 4: p[s] = exp(e[s] - gmax) -> d_out; per-block partial sums.
// ---------------------------------------------------------------------------
__global__ __launch_bounds__(256)
void k_exp(const float* __restrict__ e_in,
           const float* __restrict__ gmax,
           float* __restrict__ out,
           float* __restrict__ bsum) {
    __shared__ float sd[256];
    const int t = threadIdx.x;
    const int s = blockIdx.x * 256 + t;
    const float p = __expf(e_in[s] - *gmax);
    out[s] = p;
    sd[t] = p;
    __syncthreads();
    for (int k = 128; k > 0; k >>= 1) {
        if (t < k) sd[t] += sd[t + k];
        __syncthreads();
    }
    if (t == 0) bsum[blockIdx.x] = sd[0];
}

// ---------------------------------------------------------------------------
// Kernel 5: total = sum(bsum); out[s] *= 1/total.
// ---------------------------------------------------------------------------
__global__ __launch_bounds__(256)
void k_scale(const float* __restrict__ bsum, float* __restrict__ out) {
    __shared__ float sd[256];
    const int t = threadIdx.x;
    sd[t] = (t < NB_SOFT) ? bsum[t] : 0.f;
    __syncthreads();
    for (int k = 128; k > 0; k >>= 1) {
        if (t < k) sd[t] += sd[t + k];
        __syncthreads();
    }
    __syncthreads();
    const float inv = 1.0f / sd[0];
    const int s = blockIdx.x * 256 + t;
    out[s] *= inv;
}

// ---------------------------------------------------------------------------
extern "C" void kernel_launch(void* const* d_in, const int* in_sizes, int n_in,
                              void* d_out, int out_size, void* d_ws, size_t ws_size,
                              hipStream_t stream) {
    const float* hidden = (const float*)d_in[0];   // (1, 1024)
    const float* enc    = (const float*)d_in[1];   // (32768, 1, 1024)
    const float* W      = (const float*)d_in[2];   // (1024, 1024)
    // d_in[3] (b) intentionally unused: uniform shift, softmax-invariant.
    float* out = (float*)d_out;                    // (1,1,32768) f32

    char* ws = (char*)d_ws;
    float* v    = (float*)(ws + WS_V_OFF);
    float* e    = (float*)(ws + WS_E_OFF);
    float* bmax = (float*)(ws + WS_BMAX_OFF);
    float* gmax = (float*)(ws + WS_GMAX_OFF);
    float* bsum = (float*)(ws + WS_BSUM_OFF);

    k_v_wmma<<<HDIM / 16, 32, 0, stream>>>(hidden, W, v);
    k_energy<<<NB_ENERGY, 256, 0, stream>>>(enc, v, e, bmax);
    k_gmax<<<1, 256, 0, stream>>>(bmax, gmax);
    k_exp<<<NB_SOFT, 256, 0, stream>>>(e, gmax, out, bsum);
    k_scale<<<NB_SOFT, 256, 0, stream>>>(bsum, out);
}